// RoiPooling_7550552506676
// MI455X (gfx1250) — compile-verified
//
#include <hip/hip_runtime.h>
#include <stdint.h>

// ---------------------------------------------------------------------------
// ROI-Align (crop_and_resize, bilinear, extrapolation 0) + 2x2 max pool.
//   features : [1, 200, 200, 256] f32   (H,W,C row = 1KB contiguous per pixel)
//   rois     : [1, 1000, 4] i32 (x1,y1,x2,y2 raw; /STRIDES then normalized)
//   out      : [1, 1000, 7, 7, 256] f32
// Strategy: 1 block per ROI, 64 threads (wave32 x2), thread t owns 4 channels.
// Per pooled cell (49 per ROI): 4 bilinear samples x 4 taps = 16 x 1KB rows
// staged into LDS by the CDNA5 async copy engine (GLOBAL_LOAD_ASYNC_TO_LDS),
// double-buffered so the next cell's gathers overlap the current cell's blend.
// ---------------------------------------------------------------------------

#define POOL      7
#define GRID_S    14
#define FH        200
#define FW        200
#define FC        256
#define ROW_BYTES (FC * 4)          // 1024 B per (y,x) channel row
#define CELLS     (POOL * POOL)     // 49
#define BUF_FLOATS (16 * FC)        // 16 rows * 256 f32 = 16KB
#define BUF_BYTES  (BUF_FLOATS * 4)

// Compile-time reciprocal of the grid denominator (S-1): turns the per-sample
// grid division into a single multiply (constant-folded by the compiler).
#define GRID_RCP  (1.0f / (float)(GRID_S - 1))

// Per-cell geometry: 16 tap-row byte offsets + 16 bilinear weights
// (4 samples x 4 taps, extrapolation mask folded into the weights).
__device__ __forceinline__ void cell_geom(float sy1Hf, float sx1Wf,
                                          float dyHf,  float dxWf,
                                          int cell, int* rowOff, float* wgt)
{
    const float Hf = (float)(FH - 1);
    const float Wf = (float)(FW - 1);
    const int py = cell / POOL;
    const int px = cell - py * POOL;
#pragma unroll
    for (int s = 0; s < 4; ++s) {
        const int gy = 2 * py + (s >> 1);
        const int gx = 2 * px + (s & 1);
        const float ggy = (float)gy * GRID_RCP;
        const float ggx = (float)gx * GRID_RCP;
        const float ys = sy1Hf + ggy * dyHf;
        const float xs = sx1Wf + ggx * dxWf;
        const float y0f = floorf(ys);
        const float x0f = floorf(xs);
        const float ly = ys - y0f;
        const float lx = xs - x0f;
        const int y0 = min(max((int)y0f, 0), FH - 1);
        const int x0 = min(max((int)x0f, 0), FW - 1);
        const int y1 = min(y0 + 1, FH - 1);
        const int x1 = min(x0 + 1, FW - 1);
        const bool valid = (ys >= 0.0f) && (ys <= Hf) && (xs >= 0.0f) && (xs <= Wf);
        const float m = valid ? 1.0f : 0.0f;
        const float omly = 1.0f - ly, omlx = 1.0f - lx;
        wgt[s * 4 + 0] = m * omly * omlx;   // (y0,x0)
        wgt[s * 4 + 1] = m * omly * lx;     // (y0,x1)
        wgt[s * 4 + 2] = m * ly * omlx;     // (y1,x0)
        wgt[s * 4 + 3] = m * ly * lx;       // (y1,x1)
        rowOff[s * 4 + 0] = (y0 * FW + x0) * ROW_BYTES;
        rowOff[s * 4 + 1] = (y0 * FW + x1) * ROW_BYTES;
        rowOff[s * 4 + 2] = (y1 * FW + x0) * ROW_BYTES;
        rowOff[s * 4 + 3] = (y1 * FW + x1) * ROW_BYTES;
    }
}

// Issue 16 async global->LDS 128-bit copies per thread: 64 threads move
// 16 rows x 1KB into the LDS buffer. GVS mode: saddr = feature base (SGPR
// pair), vaddr = per-lane byte offset, dest VGPR = LDS byte address.
__device__ __forceinline__ void stage_cell(uint64_t base, unsigned ldsDstBase,
                                           const int* rowOff, int t)
{
#pragma unroll
    for (int k = 0; k < 16; ++k) {
        const unsigned goff = (unsigned)rowOff[k] + (unsigned)(t * 16);
        const unsigned dst  = ldsDstBase + (unsigned)(k * ROW_BYTES) + (unsigned)(t * 16);
        asm volatile("global_load_async_to_lds_b128 %0, %1, %2"
                     :: "v"(dst), "v"(goff), "s"(base) : "memory");
    }
}

__global__ __launch_bounds__(64) void roi_align_maxpool_kernel(
    const int* __restrict__ image_shape,
    const float* __restrict__ feat,
    const int* __restrict__ rois,
    float* __restrict__ out,
    int nroi)
{
    __shared__ __align__(16) float smem[2 * BUF_FLOATS];   // 32 KB, double buffer

    const int t = (int)threadIdx.x;   // 0..63 -> channels [4t, 4t+4)
    const int n = (int)blockIdx.x;    // ROI index
    if (n >= nroi) return;

    const float hh = (float)image_shape[0];
    const float ww = (float)image_shape[1];
    const int4 r = ((const int4*)rois)[n];            // x1,y1,x2,y2 (raw)
    const float sx1 = ((float)r.x * 0.25f) / ww;
    const float sy1 = ((float)r.y * 0.25f) / hh;
    const float sx2 = ((float)r.z * 0.25f) / ww;
    const float sy2 = ((float)r.w * 0.25f) / hh;

    // Hoist ROI-constant terms of the sample-coordinate affine map.
    const float Hf = (float)(FH - 1);
    const float Wf = (float)(FW - 1);
    const float sy1Hf = sy1 * Hf;
    const float sx1Wf = sx1 * Wf;
    const float dyHf  = (sy2 - sy1) * Hf;
    const float dxWf  = (sx2 - sx1) * Wf;

    const uint64_t featBase = (uint64_t)(uintptr_t)feat;
    // Low 32 bits of the flat shared-pointer = workgroup-relative LDS offset.
    const unsigned ldsBase = (unsigned)(uintptr_t)(&smem[0]);

    int   rowOff[16], rowOffN[16];
    float wgt[16],    wgtN[16];

    // Prime the pipeline with cell 0.
    cell_geom(sy1Hf, sx1Wf, dyHf, dxWf, 0, rowOff, wgt);
    stage_cell(featBase, ldsBase, rowOff, t);

    for (int cell = 0; cell < CELLS; ++cell) {
        // Issue next cell into the other buffer while current is in flight.
        if (cell + 1 < CELLS) {
            cell_geom(sy1Hf, sx1Wf, dyHf, dxWf, cell + 1, rowOffN, wgtN);
            const unsigned nxtBuf = ldsBase + ((cell & 1) ? 0u : (unsigned)BUF_BYTES);
            stage_cell(featBase, nxtBuf, rowOffN, t);
            // 32 outstanding max; oldest 16 (current cell) must be done.
            asm volatile("s_wait_asynccnt 16" ::: "memory");
        } else {
            asm volatile("s_wait_asynccnt 0" ::: "memory");
        }
        __syncthreads();   // all waves' current-cell rows are now in LDS

        const int fb = (cell & 1) ? BUF_FLOATS : 0;
        float4 best;
#pragma unroll
        for (int s = 0; s < 4; ++s) {
            const float4 v0 = *(const float4*)&smem[fb + (s * 4 + 0) * FC + t * 4];
            const float4 v1 = *(const float4*)&smem[fb + (s * 4 + 1) * FC + t * 4];
            const float4 v2 = *(const float4*)&smem[fb + (s * 4 + 2) * FC + t * 4];
            const float4 v3 = *(const float4*)&smem[fb + (s * 4 + 3) * FC + t * 4];
            const float w0 = wgt[s * 4 + 0], w1 = wgt[s * 4 + 1];
            const float w2 = wgt[s * 4 + 2], w3 = wgt[s * 4 + 3];
            float4 a;
            a.x = w0 * v0.x + w1 * v1.x + w2 * v2.x + w3 * v3.x;
            a.y = w0 * v0.y + w1 * v1.y + w2 * v2.y + w3 * v3.y;
            a.z = w0 * v0.z + w1 * v1.z + w2 * v2.z + w3 * v3.z;
            a.w = w0 * v0.w + w1 * v1.w + w2 * v2.w + w3 * v3.w;
            if (s == 0) {
                best = a;
            } else {
                best.x = fmaxf(best.x, a.x);
                best.y = fmaxf(best.y, a.y);
                best.z = fmaxf(best.z, a.z);
                best.w = fmaxf(best.w, a.w);
            }
        }
        float* op = out + ((size_t)n * CELLS + cell) * FC + (size_t)(t * 4);
        *(float4*)op = best;

        __syncthreads();   // everyone done reading this buffer before reuse

#pragma unroll
        for (int k = 0; k < 16; ++k) { rowOff[k] = rowOffN[k]; wgt[k] = wgtN[k]; }
    }
}

extern "C" void kernel_launch(void* const* d_in, const int* in_sizes, int n_in,
                              void* d_out, int out_size, void* d_ws, size_t ws_size,
                              hipStream_t stream) {
    const int*   image_shape = (const int*)d_in[0];   // [2] i32
    const float* feat        = (const float*)d_in[1]; // [1,200,200,256] f32
    const int*   rois        = (const int*)d_in[2];   // [1,1000,4] i32
    float*       out         = (float*)d_out;         // [1,1000,7,7,256] f32

    const int nroi = in_sizes[2] / 4;                 // 1000
    roi_align_maxpool_kernel<<<dim3(nroi), dim3(64), 0, stream>>>(
        image_shape, feat, rois, out, nroi);
}